// IRadon_51539607552702
// MI455X (gfx1250) — compile-verified
//
#include <hip/hip_runtime.h>

// ---------------------------------------------------------------------------
// FBP (inverse Radon) for x:[4,1,512,180] -> out:[4,1,512,512], fp32.
//
// Stage 1: ramp filtering as a dense GEMM  F = M @ X
//   M[d,k] = h(d-k), h(0)=0.5, h(odd delta)=-2/(pi^2 delta^2), h(even)=0
//   (closed form of ifft(2*Re(fft(ramp_kernel))) incl. the zero-pad wrap).
//   A fragments are generated analytically in registers (zero memory traffic)
//   using branchless selects + v_rcp_f32 so the VALU work co-executes with
//   V_WMMA_F32_16X16X4_F32 instead of starving it.
//   X is [512 det, 720 cols], col = b*180 + a. One wave per 16x16 tile,
//   128 k-steps. Result stored transposed: F[col*512 + d].
//
// Stage 2: back-projection. 1 thread/pixel, 180-angle loop, 4 batches share
//   the interpolation index/weights; sin/cos staged in LDS. All buffers are
//   L2-resident on MI455X (192MB L2) -> latency-bound, WMMA carries the math.
// ---------------------------------------------------------------------------

typedef float v2f __attribute__((ext_vector_type(2)));
typedef float v8f __attribute__((ext_vector_type(8)));

#define N_DET   512
#define N_ANG   180
#define N_B     4
#define NCOL    (N_B * N_ANG)   // 720
#define IMG     512
#define TILES_D (N_DET / 16)    // 32
#define TILES_C (NCOL / 16)     // 45

// Branchless, division-free ramp kernel:
//   h(0) = 0.5 ; h(odd d) = -2/(pi^2 d^2) ; h(even d != 0) = 0
// v_rcp_f32 (TRANS) + 2 v_cndmask; no exec-mask branches, no IEEE divide.
__device__ __forceinline__ float ramp_h(int delta) {
    const float fd = (float)delta;
    const float r  = __builtin_amdgcn_rcpf(fd * fd);       // v_rcp_f32
    float v = (delta & 1) ? (-0.20264236728467555f * r) : 0.0f;
    return (delta == 0) ? 0.5f : v;
}

__global__ void __launch_bounds__(128)
fbp_filter_gemm(const float* __restrict__ x, float* __restrict__ F) {
    const int wave    = (blockIdx.x * blockDim.x + threadIdx.x) >> 5; // 0..1439
    const int lane    = threadIdx.x & 31;
    const int tile_d  = wave % TILES_D;
    const int tile_c  = wave / TILES_D;          // 0..44
    const int dBase   = tile_d * 16;
    const int colBase = tile_c * 16;

    const int halfsel = lane >> 4;               // 0: K pair {0,1}, 1: {2,3}
    const int lane16  = lane & 15;

    const int d   = dBase + lane16;              // A-matrix row (M) for this lane
    const int col = colBase + lane16;            // B/C/D column (N) for this lane
    const int b   = col / N_ANG;
    const int a   = col - b * N_ANG;
    // x[b, 0, k, a] = x[(b*512 + k)*180 + a]
    const float* __restrict__ xb = x + (size_t)b * N_DET * N_ANG + a;

    v8f c = {0.f, 0.f, 0.f, 0.f, 0.f, 0.f, 0.f, 0.f};

    // delta for av.x at k-step: d - (k + halfsel*2); decreases by 4 per step
    int delta = d - halfsel * 2;

#pragma unroll 4
    for (int k = 0; k < N_DET; k += 4) {
        // A fragment: analytic filter-matrix entries, registers only.
        v2f av;
        av.x = ramp_h(delta);
        av.y = ramp_h(delta - 1);
        delta -= 4;
        // B fragment: rows K=kk,kk+1 of column `col` of X.
        const int kk = k + halfsel * 2;
        v2f bv;
        bv.x = xb[(size_t)kk * N_ANG];
        bv.y = xb[(size_t)(kk + 1) * N_ANG];
        c = __builtin_amdgcn_wmma_f32_16x16x4_f32(
                /*neg_a=*/false, av, /*neg_b=*/false, bv,
                /*c_mod=*/(short)0, c, /*reuse_a=*/false, /*reuse_b=*/false);
    }

    // C/D layout: VGPR r, lanes 0-15 -> M=r, lanes 16-31 -> M=r+8, N=lane16.
    // Store transposed: F[col*512 + d]
    float* __restrict__ Fc = F + (size_t)col * N_DET + dBase + halfsel * 8;
#pragma unroll
    for (int r = 0; r < 8; ++r) Fc[r] = c[r];
}

__global__ void __launch_bounds__(256)
fbp_backproject(const float* __restrict__ F, float* __restrict__ out) {
    __shared__ float s_cos[N_ANG];
    __shared__ float s_sin[N_ANG];
    if (threadIdx.x < N_ANG) {
        float th = (float)threadIdx.x * (3.14159265358979323846f / 180.0f);
        s_cos[threadIdx.x] = cosf(th);
        s_sin[threadIdx.x] = sinf(th);
    }
    __syncthreads();

    const int pix = blockIdx.x * blockDim.x + threadIdx.x;   // 0..262143
    const int px  = pix & (IMG - 1);
    const int py  = pix >> 9;
    // linspace(-1,1,512), meshgrid 'ij': row -> y, col -> x
    const float xg = -1.0f + 2.0f * (float)px * (1.0f / 511.0f);
    const float yg = -1.0f + 2.0f * (float)py * (1.0f / 511.0f);

    float acc[N_B] = {0.f, 0.f, 0.f, 0.f};

    for (int a = 0; a < N_ANG; ++a) {
        const float t   = xg * s_cos[a] - yg * s_sin[a];
        const float iy  = (t + 1.0f) * 0.5f * (float)(N_DET - 1);
        const float i0f = floorf(iy);
        const float w   = iy - i0f;
        const int   i0  = (int)i0f;
        const int   i1  = i0 + 1;
        const bool  v0  = (i0 >= 0) && (i0 < N_DET);
        const bool  v1  = (i1 >= 0) && (i1 < N_DET);
        const int   c0  = min(max(i0, 0), N_DET - 1);
        const int   c1  = min(max(i1, 0), N_DET - 1);
        const float w0  = 1.0f - w;
#pragma unroll
        for (int b = 0; b < N_B; ++b) {
            const float* colp = F + (size_t)(b * N_ANG + a) * N_DET;
            const float f0 = v0 ? colp[c0] : 0.0f;
            const float f1 = v1 ? colp[c1] : 0.0f;
            acc[b] += w0 * f0 + w * f1;
        }
    }

    const bool inside = (xg * xg + yg * yg) <= 1.0f;
    const float scale = inside ? (3.14159265358979323846f / (2.0f * (float)N_ANG))
                               : 0.0f;
#pragma unroll
    for (int b = 0; b < N_B; ++b) {
        out[(size_t)b * IMG * IMG + (size_t)py * IMG + px] = acc[b] * scale;
    }
}

extern "C" void kernel_launch(void* const* d_in, const int* in_sizes, int n_in,
                              void* d_out, int out_size, void* d_ws, size_t ws_size,
                              hipStream_t stream) {
    const float* x = (const float*)d_in[0];   // [4,1,512,180] fp32
    float* F   = (float*)d_ws;                // filtered sinogram [720][512] fp32
    float* out = (float*)d_out;               // [4,1,512,512] fp32

    // Stage 1: WMMA GEMM — 32x45 = 1440 tiles, 4 waves (128 threads) / block
    fbp_filter_gemm<<<(TILES_D * TILES_C) / 4, 128, 0, stream>>>(x, F);
    // Stage 2: back-projection — one thread per pixel
    fbp_backproject<<<(IMG * IMG) / 256, 256, 0, stream>>>(F, out);
}